// ShieldLayer_40467181863380
// MI455X (gfx1250) — compile-verified
//
#include <hip/hip_runtime.h>
#include <hip/hip_bf16.h>

// CDNA5 / gfx1250: one wave32 per 16-row batch tile, private LDS copy of the
// augmented prediction row-block, sequential scan over 512 variables using
// exact-precision V_WMMA_F32_16X16X4_F32.
//
// Operand roles: D[c][row] = sum_k W[c][k] * P[row][k]
//   A (16x4, M=constraint) <- global b64 loads of pos/neg_mats[v]
//   B (4x16, N=batch row)  <- LDS b64 loads of P (stride 516 -> conflict-free)
// D layout: lane L holds row = L%16, constraints c = d + 8*(L/16) in VGPR d,
// so the max/min over the 16 constraints is 7 in-lane ops + one xor-16 shuffle,
// and the clamp of P[:,v] is a single 16-lane ds_load/ds_store.

typedef __attribute__((ext_vector_type(2))) float v2f;
typedef __attribute__((ext_vector_type(8))) float v8f;

#define NVAR  512            // V
#define NCON  16             // constraints per variable
#define NN1   513            // V + 1 (bias column)
#define PSTR  516            // padded LDS row stride (dwords); 516%64==4 -> conflict-free
#define TROWS 16             // batch rows per wave

__global__ __launch_bounds__(32)
void shield_scan_kernel(const float* __restrict__ preds,
                        const float* __restrict__ pos_mats,
                        const float* __restrict__ neg_mats,
                        float* __restrict__ out)
{
    __shared__ float P[TROWS * PSTR];      // 33,024 B per (single-wave) workgroup

    const int lane = (int)threadIdx.x;     // 0..31 (wave32)
    const int half = lane >> 4;            // 0 | 1
    const int nc   = lane & 15;            // batch-row id (B/D col) and W row id
    const long row0 = (long)blockIdx.x * TROWS;

    // ---- load this tile's predictions into LDS, append bias column = 1.0 ----
    for (int i = lane; i < TROWS * NVAR; i += 32) {
        const int r = i >> 9;              // i / 512
        const int c = i & (NVAR - 1);      // i % 512
        P[r * PSTR + c] = preds[(row0 + r) * NVAR + c];
    }
    if (lane < TROWS) P[lane * PSTR + NVAR] = 1.0f;
    // single wave owns this LDS slice; DS ops are in-order within a wave.

    // B-fragment base: lane L reads P[row = nc][k + 2*half .. +1]
    const int boff = nc * PSTR + 2 * half;

    for (int v = 0; v < NVAR; ++v) {
        const float* __restrict__ pm = pos_mats + (size_t)v * NCON * NN1;
        const float* __restrict__ nm = neg_mats + (size_t)v * NCON * NN1;
        // pull next step's matrices toward L2/WGP$ (global_prefetch_b8)
        __builtin_prefetch(pm + NCON * NN1, 0, 1);
        __builtin_prefetch(nm + NCON * NN1, 0, 1);

        // A-fragment base: lane L reads mat[v][c = nc][k + 2*half .. +1]
        const float* __restrict__ pmr = pm + nc * NN1 + 2 * half;
        const float* __restrict__ nmr = nm + nc * NN1 + 2 * half;

        // Bias column (p[:,512] == 1.0 always): fold into accumulator init.
        // Lane L holds the bias weight for constraint c = nc; D wants
        // acc[d] = bias(c = d + 8*half) -> broadcast via shuffles.
        const float bpl = pm[nc * NN1 + NVAR];
        const float bnl = nm[nc * NN1 + NVAR];
        v8f accP, accN;
#pragma unroll
        for (int d = 0; d < 8; ++d) {
            accP[d] = __shfl(bpl, d + 8 * half, 32);
            accN[d] = __shfl(bnl, d + 8 * half, 32);
        }

        // Acyclic mask: matrix cols in [v, 512) are zero, so rounding K up to a
        // multiple of 4 needs no tail masking.
        const int kmax = (v + 3) & ~3;
        for (int k = 0; k < kmax; k += 4) {
            v2f b, wp, wn;
            b.x  = P[boff + k];  b.y  = P[boff + k + 1];   // ds_load (LDS)
            wp.x = pmr[k];       wp.y = pmr[k + 1];        // global (L2-resident)
            wn.x = nmr[k];       wn.y = nmr[k + 1];
            // D = W(16x4 f32) * P^T(4x16 f32) + C ; two independent chains
            accP = __builtin_amdgcn_wmma_f32_16x16x4_f32(
                       false, wp, false, b, (short)0, accP, false, false);
            accN = __builtin_amdgcn_wmma_f32_16x16x4_f32(
                       false, wn, false, b, (short)0, accN, false, false);
        }

        // Reduce over constraints: 8 in-lane (c = 8*half..8*half+7), then the
        // other half via one xor-16 shuffle. Result: every lane holds the final
        // lower/upper bound for batch row = nc.
        float lo = accP[0], hi = accN[0];
#pragma unroll
        for (int d = 1; d < 8; ++d) {
            lo = fmaxf(lo, accP[d]);
            hi = fminf(hi, accN[d]);
        }
        lo = fmaxf(lo, __shfl_xor(lo, 16, 32));
        hi = fminf(hi, __shfl_xor(hi, 16, 32));

        // Clamp p[:,v] into [lower, upper] — 16 lanes in parallel.
        if (half == 0) {
            const float cur = P[nc * PSTR + v];
            P[nc * PSTR + v] = fminf(fmaxf(cur, lo), hi);
        }
    }

    // ---- write corrected predictions (drop bias column) ----
    for (int i = lane; i < TROWS * NVAR; i += 32) {
        const int r = i >> 9;
        const int c = i & (NVAR - 1);
        out[(row0 + r) * NVAR + c] = P[r * PSTR + c];
    }
}

extern "C" void kernel_launch(void* const* d_in, const int* in_sizes, int n_in,
                              void* d_out, int out_size, void* d_ws, size_t ws_size,
                              hipStream_t stream) {
    (void)in_sizes; (void)n_in; (void)out_size; (void)d_ws; (void)ws_size;
    const float* preds    = (const float*)d_in[0];   // [8192, 512]
    const float* pos_mats = (const float*)d_in[1];   // [512, 16, 513]
    const float* neg_mats = (const float*)d_in[2];   // [512, 16, 513]
    float* out = (float*)d_out;                      // [8192, 512]

    const int tiles = 8192 / TROWS;                  // 512 single-wave workgroups
    shield_scan_kernel<<<dim3(tiles), dim3(32), 0, stream>>>(preds, pos_mats, neg_mats, out);
}